// kernel_generated_3_62904091017337
// MI455X (gfx1250) — compile-verified
//
#include <hip/hip_runtime.h>

// Shapes (hard-coded by the reference)
#define Bb   128   // batch
#define C1c  128   // channel split 1 (reduction group)
#define C2c  14    // channel split 2 (rolled axis)
#define Ss   14    // spatial
#define Kk   3     // conv kernel taps
#define Hh   256   // hidden
#define Ee   2     // expand
#define KRED (C1c * Kk)   // 384 reduction length, reordered as ki' = i*128 + j

// LDS layout (floats)
#define XS_FLOATS (C2c * 16 * C1c)  // x[b] staged as xs[n][t][j], zero pad t=0,15 : 28672
#define WB_FLOATS (KRED * 16)       // w1 panel as wb[pair][h_local][2]           :  6144
#define TT_FLOATS (C2c * 16 * 16)   // t4 tiles [n][m(0..15)][h_local]            :  3584

typedef __attribute__((ext_vector_type(2))) float v2f;
typedef __attribute__((ext_vector_type(8))) float v8f;

extern "C" __global__ __launch_bounds__(256)
void fused_groupconv_mix_wmma(const float* __restrict__ x,
                              const float* __restrict__ w0,
                              const float* __restrict__ w1,
                              const int*   __restrict__ shift_p,
                              float*       __restrict__ out)
{
    extern __shared__ float smem[];
    float* xs = smem;                           // [C2][16][C1]
    float* wb = smem + XS_FLOATS;               // [KRED/2][16][2]
    float* tt = smem + XS_FLOATS + WB_FLOATS;   // [C2][16][16]

    const int tid = threadIdx.x;
    const int blk = blockIdx.x;
    const int b   = blk >> 4;          // batch index
    const int h0  = (blk & 15) << 4;   // H-tile base

    const float* xb = x + (size_t)b * (C1c * C2c * Ss);

    // gfx1250 prefetch into the cache hierarchy ahead of the staging loops
    __builtin_prefetch(xb + tid * 98, 0, 1);          // spans the 25088-float x[b] slice
    __builtin_prefetch(w1 + tid * 384, 0, 1);         // spans the 98304-float w1 tensor

    // ---------------- stage x[b] into LDS: xs[n][t][j], spatial zero padding ----------------
    for (int idx = tid; idx < C2c * 2 * C1c; idx += 256) {
        int j = idx & 127;
        int r = idx >> 7;              // 0..27
        int n = r >> 1;
        int t = (r & 1) ? 15 : 0;
        xs[(n * 16 + t) * C1c + j] = 0.0f;
    }
    for (int g = tid; g < C1c * C2c * Ss; g += 256) {   // contiguous global reads
        int j = g / (C2c * Ss);
        int r = g - j * (C2c * Ss);
        int n = r / Ss;
        int m = r - n * Ss;
        xs[(n * 16 + (m + 1)) * C1c + j] = xb[g];
    }
    // ---------------- stage w1 H-tile panel: wb[ki'>>1][hl][ki'&1], ki' = i*128 + j ----------------
    for (int idx = tid; idx < KRED * 16; idx += 256) {
        int hl  = idx & 15;
        int kip = idx >> 4;            // ki' in [0,384)
        int j   = kip & 127;
        int i   = kip >> 7;
        wb[(kip >> 1) * 32 + hl * 2 + (kip & 1)] = w1[(j * Kk + i) * Hh + h0 + hl];
    }
    __syncthreads();

    // ---------------- WMMA GEMM: t4[n][m][h] over reordered K = 384 ----------------
    const int wave  = tid >> 5;
    const int lane  = tid & 31;
    const int ml    = lane & 15;       // A-matrix row m / B-N h_local / C column lane
    const int khalf = lane >> 4;       // K pair {0,1} vs {2,3} inside a K=4 block
    const int mm    = (ml < Ss) ? ml : 0;   // clamp pad rows 14,15 (their C rows are never read)

    const int  n0  = wave;             // first n tile
    const int  n1  = wave + 8;         // second n tile (may be >= 14)
    const bool two = (n1 < C2c);
    const int  n1c = two ? n1 : 0;     // computed unconditionally, stored conditionally

    v8f acc0 = {0.f,0.f,0.f,0.f,0.f,0.f,0.f,0.f};
    v8f acc1 = acc0;

    #pragma unroll
    for (int i = 0; i < Kk; ++i) {
        const int t = mm + i;                                     // <= 15, in-row
        const float* pa0 = xs + ((n0  * 16 + t) * C1c + 2 * khalf);   // 8B-aligned
        const float* pa1 = xs + ((n1c * 16 + t) * C1c + 2 * khalf);
        const float* pb  = wb + ((64 * i + khalf) * 32 + ml * 2);

        #pragma unroll 8
        for (int jb = 0; jb < 32; ++jb) {                         // j = 4*jb + 2*khalf (+1)
            v2f af0 = *(const v2f*)(pa0 + jb * 4);                // ds_load_b64
            v2f af1 = *(const v2f*)(pa1 + jb * 4);                // ds_load_b64
            v2f bf  = *(const v2f*)(pb  + jb * 64);               // ds_load_b64
            acc0 = __builtin_amdgcn_wmma_f32_16x16x4_f32(
                       false, af0, false, bf, (short)0, acc0, false, false);
            acc1 = __builtin_amdgcn_wmma_f32_16x16x4_f32(
                       false, af1, false, bf, (short)0, acc1, false, false);
        }
    }

    // ---------------- park C tiles in LDS: tt[n][Mrow][h_local] ----------------
    // C layout: lanes 0-15 -> N=lane, Mrow = r; lanes 16-31 -> N=lane-16, Mrow = r+8
    {
        const int mbase = khalf << 3;
        #pragma unroll
        for (int r = 0; r < 8; ++r) {
            tt[(n0 * 16 + (mbase + r)) * 16 + ml] = acc0[r];
        }
        if (two) {
            #pragma unroll
            for (int r = 0; r < 8; ++r) {
                tt[(n1 * 16 + (mbase + r)) * 16 + ml] = acc1[r];
            }
        }
    }
    __syncthreads();

    // ---------------- epilogue: both rolls + w0 expand/sum, transposed store ----------------
    // out[b,h,n,m] = sum_e t4[b, (q/2 - shift)%14, m, h] * w0[h, q&1],  q = (2n+e-shift)%28
    const int shift = *shift_p;
    for (int idx = tid; idx < 16 * C2c * Ss; idx += 256) {
        const int m  = idx % Ss;
        const int n  = (idx / Ss) % C2c;
        const int hl = idx / (Ss * C2c);
        const int h  = h0 + hl;
        float acc = 0.0f;
        #pragma unroll
        for (int e = 0; e < Ee; ++e) {
            int q = (2 * n + e - shift) % (C2c * Ee);
            q = (q + C2c * Ee) % (C2c * Ee);
            const int nh = q >> 1;
            const int eh = q & 1;
            int ns = (nh - shift) % C2c;
            ns = (ns + C2c) % C2c;
            acc += tt[(ns * 16 + m) * 16 + hl] * w0[h * Ee + eh];
        }
        out[(((size_t)b * Hh + h) * C2c + n) * Ss + m] = acc;
    }
}

extern "C" void kernel_launch(void* const* d_in, const int* in_sizes, int n_in,
                              void* d_out, int out_size, void* d_ws, size_t ws_size,
                              hipStream_t stream)
{
    const float* x     = (const float*)d_in[0];
    const float* w0    = (const float*)d_in[1];
    const float* w1    = (const float*)d_in[2];
    const int*   shift = (const int*)  d_in[3];
    float*       out   = (float*)d_out;

    const size_t smem_bytes = (size_t)(XS_FLOATS + WB_FLOATS + TT_FLOATS) * sizeof(float); // ~150 KB < 320 KB
    dim3 grid(Bb * 16);   // 128 batches x 16 H-tiles
    dim3 block(256);      // 8 wave32s
    hipLaunchKernelGGL(fused_groupconv_mix_wmma, grid, block, smem_bytes, stream,
                       x, w0, w1, shift, out);
}